// GCN3D_37873021616798
// MI455X (gfx1250) — compile-verified
//
#include <hip/hip_runtime.h>
#include <hip/hip_bf16.h>

typedef float v2f __attribute__((ext_vector_type(2)));
typedef float v8f __attribute__((ext_vector_type(8)));

#define NEI 32
#define POOL_NEI 4
#define VSTRIDE (4096 * 3)   // batch stride of the original vertex array
#define NEG_BIG (-3.402823466e38f)

// ---------------------------------------------------------------------------
// query_ball: first-K in-radius indices in ascending order, padded with first.
// ---------------------------------------------------------------------------
__global__ void query_ball_kernel(const float* __restrict__ verts,
                                  int npts, int nq, float r2, int K,
                                  int* __restrict__ idx)
{
    int q = blockIdx.x * blockDim.x + threadIdx.x;
    int b = blockIdx.y;
    if (q >= nq) return;
    const float* vb = verts + (size_t)b * VSTRIDE;
    float qx = vb[q * 3 + 0], qy = vb[q * 3 + 1], qz = vb[q * 3 + 2];
    int* out = idx + ((size_t)b * nq + q) * K;
    int cnt = 0;
    int first = q;  // self is always within radius (d==0)
    for (int p = 0; p < npts; ++p) {
        float dx = vb[p * 3 + 0] - qx;
        float dy = vb[p * 3 + 1] - qy;
        float dz = vb[p * 3 + 2] - qz;
        float d = dx * dx + dy * dy + dz * dz;
        if (d <= r2) {
            if (cnt == 0) first = p;
            out[cnt++] = p;
            if (cnt >= K) break;
        }
    }
    for (int j = cnt; j < K; ++j) out[j] = first;
}

// ---------------------------------------------------------------------------
// Normalize direction matrix columns: sd[:,m] = d[:,m] / (||d[:,m]|| + eps)
// ---------------------------------------------------------------------------
__global__ void norm_dirs_kernel(const float* __restrict__ d,
                                 float* __restrict__ sd, int M)
{
    int m = blockIdx.x * blockDim.x + threadIdx.x;
    if (m >= M) return;
    float x = d[m], y = d[M + m], z = d[2 * M + m];
    float inv = 1.0f / (sqrtf(x * x + y * y + z * z) + 1e-12f);
    sd[m] = x * inv;
    sd[M + m] = y * inv;
    sd[2 * M + m] = z * inv;
}

// ---------------------------------------------------------------------------
// Wide WMMA GEMM: each wave computes a full 16x64 strip (4 x 16x16 tiles),
// straight-line, no per-subtile conditionals. One A fragment feeds 4
// back-to-back V_WMMA_F32_16X16X4_F32 per K-step. The only guard is a
// scalar branch (wave index via readfirstlane) so EXEC is never masked
// around a WMMA (ISA 7.12: EXEC must be all-ones).
// ---------------------------------------------------------------------------
__global__ void wmma_gemm_bias_wide(const float* __restrict__ A, int lda,
                                    const float* __restrict__ W, int N,
                                    const float* __restrict__ bias,
                                    float* __restrict__ C, int K, int strips)
{
    const int lane = threadIdx.x & 31;
    const int lid = lane & 15;
    const int hi = lane >> 4;
    const int wave = __builtin_amdgcn_readfirstlane((int)(threadIdx.x >> 5));
    const int strip = blockIdx.x * (blockDim.x >> 5) + wave;
    if (strip >= strips) return;      // scalar cmp -> s_cbranch, EXEC untouched
    const int col0 = strip * 64;
    const int row0 = blockIdx.y * 16;

    const float bv0 = bias[col0 + 0 + lid];
    const float bv1 = bias[col0 + 16 + lid];
    const float bv2 = bias[col0 + 32 + lid];
    const float bv3 = bias[col0 + 48 + lid];
    v8f c0, c1, c2, c3;
#pragma unroll
    for (int r = 0; r < 8; ++r) { c0[r] = bv0; c1[r] = bv1; c2[r] = bv2; c3[r] = bv3; }

    // A fragment: lane lid holds row (row0+lid), K offsets {2*hi, 2*hi+1}
    const float* arow = A + (size_t)(row0 + lid) * lda + 2 * hi;
    // B fragments: lane lid holds col (col0+16j+lid), K offsets {2*hi, 2*hi+1}
    const float* wbase = W + (size_t)(2 * hi) * N + col0 + lid;

    for (int k0 = 0; k0 < K; k0 += 4) {
        v2f a;
        a.x = arow[k0];
        a.y = arow[k0 + 1];
        const float* wk = wbase + (size_t)k0 * N;
        v2f b0, b1, b2, b3;
        b0.x = wk[0];  b0.y = wk[(size_t)N + 0];
        b1.x = wk[16]; b1.y = wk[(size_t)N + 16];
        b2.x = wk[32]; b2.y = wk[(size_t)N + 32];
        b3.x = wk[48]; b3.y = wk[(size_t)N + 48];
        c0 = __builtin_amdgcn_wmma_f32_16x16x4_f32(false, a, false, b0, (short)0, c0, false, false);
        c1 = __builtin_amdgcn_wmma_f32_16x16x4_f32(false, a, false, b1, (short)0, c1, false, false);
        c2 = __builtin_amdgcn_wmma_f32_16x16x4_f32(false, a, false, b2, (short)0, c2, false, false);
        c3 = __builtin_amdgcn_wmma_f32_16x16x4_f32(false, a, false, b3, (short)0, c3, false, false);
        // Pull the next K-block of W toward the caches (global_prefetch_b8).
        if (k0 + 4 < K) __builtin_prefetch(wk + (size_t)4 * N, 0, 3);
    }

    float* crow = C + (size_t)(row0 + 8 * hi) * N + col0 + lid;
#pragma unroll
    for (int r = 0; r < 8; ++r) {
        crow[(size_t)r * N + 0]  = c0[r];
        crow[(size_t)r * N + 16] = c1[r];
        crow[(size_t)r * N + 32] = c2[r];
        crow[(size_t)r * N + 48] = c3[r];
    }
}

// ---------------------------------------------------------------------------
// Tail WMMA GEMM: one 16x16 tile per wave for the N % 64 columns.
// ---------------------------------------------------------------------------
__global__ void wmma_gemm_bias_tile(const float* __restrict__ A, int lda,
                                    const float* __restrict__ W, int N,
                                    const float* __restrict__ bias,
                                    float* __restrict__ C, int K,
                                    int colBase, int tiles)
{
    const int lane = threadIdx.x & 31;
    const int lid = lane & 15;
    const int hi = lane >> 4;
    const int wave = __builtin_amdgcn_readfirstlane((int)(threadIdx.x >> 5));
    const int tile = blockIdx.x * (blockDim.x >> 5) + wave;
    if (tile >= tiles) return;        // scalar branch
    const int col0 = colBase + tile * 16;
    const int row0 = blockIdx.y * 16;

    const float bv = bias[col0 + lid];
    v8f c;
#pragma unroll
    for (int r = 0; r < 8; ++r) c[r] = bv;

    const float* arow = A + (size_t)(row0 + lid) * lda + 2 * hi;
    const float* wcol = W + (size_t)(2 * hi) * N + col0 + lid;

    for (int k0 = 0; k0 < K; k0 += 4) {
        v2f a, bm;
        a.x = arow[k0];
        a.y = arow[k0 + 1];
        bm.x = wcol[(size_t)k0 * N];
        bm.y = wcol[(size_t)k0 * N + N];
        c = __builtin_amdgcn_wmma_f32_16x16x4_f32(false, a, false, bm,
                                                  (short)0, c, false, false);
    }

    float* crow = C + (size_t)(row0 + 8 * hi) * N + col0 + lid;
#pragma unroll
    for (int r = 0; r < 8; ++r) crow[(size_t)r * N] = c[r];
}

// ---------------------------------------------------------------------------
// Fused aggregation: out[b,v,outOff+oc] =
//   relu?( center + sum_{s<4} max_{k<32} relu(dn_k . sd[:,s*o+oc]) * supp )
// HasFeat=false -> conv_surface (supp==1, center==0).
// ---------------------------------------------------------------------------
template <bool HasFeat>
__global__ void conv_agg_kernel(const int* __restrict__ idx,
                                const float* __restrict__ verts,
                                const float* __restrict__ sd,
                                const float* __restrict__ fout,
                                float* __restrict__ out,
                                int V, int o, int outStride, int outOff,
                                int applyRelu)
{
    __shared__ float dnx[NEI], dny[NEI], dnz[NEI];
    __shared__ int nb[NEI];
    const int v = blockIdx.x;
    const int b = blockIdx.y;
    const int t = threadIdx.x;
    const float* vb = verts + (size_t)b * VSTRIDE;
    if (t < NEI) {
        int j = idx[((size_t)b * V + v) * NEI + t];
        nb[t] = j;
        float dx = vb[j * 3 + 0] - vb[v * 3 + 0];
        float dy = vb[j * 3 + 1] - vb[v * 3 + 1];
        float dz = vb[j * 3 + 2] - vb[v * 3 + 2];
        float inv = 1.0f / (sqrtf(dx * dx + dy * dy + dz * dz) + 1e-12f);
        dnx[t] = dx * inv;
        dny[t] = dy * inv;
        dnz[t] = dz * inv;
    }
    __syncthreads();

    const int M = 4 * o;        // S*o
    const int fstride = 5 * o;  // (S+1)*o
    const size_t row = (size_t)b * V + v;

    for (int oc = t; oc < o; oc += blockDim.x) {
        float acc = 0.0f;
#pragma unroll
        for (int s = 0; s < 4; ++s) {
            const int m = s * o + oc;
            const float sx = sd[m], sy = sd[M + m], sz = sd[2 * M + m];
            float best = NEG_BIG;
#pragma unroll 4
            for (int k = 0; k < NEI; ++k) {
                float th = fmaxf(dnx[k] * sx + dny[k] * sy + dnz[k] * sz, 0.0f);
                float val;
                if (HasFeat) {
                    val = th * fout[((size_t)b * V + nb[k]) * fstride + o + m];
                } else {
                    val = th;
                }
                best = fmaxf(best, val);
            }
            acc += best;
        }
        float res = HasFeat ? (fout[row * fstride + oc] + acc) : acc;
        if (applyRelu) res = fmaxf(res, 0.0f);
        out[row * outStride + outOff + oc] = res;
    }
}

// ---------------------------------------------------------------------------
// pool: out[b,v,c] = max_{j<4} feat[b, idx[b,v,j], c]   (v < Vout)
// ---------------------------------------------------------------------------
__global__ void pool_max_kernel(const float* __restrict__ feat, int C, int Vin,
                                const int* __restrict__ idx,
                                float* __restrict__ out, int outStride, int Vout)
{
    int c = blockIdx.x * blockDim.x + threadIdx.x;
    int v = blockIdx.y;
    int b = blockIdx.z;
    if (c >= C) return;
    const int* ib = idx + ((size_t)b * Vout + v) * POOL_NEI;
    float m = NEG_BIG;
#pragma unroll
    for (int j = 0; j < POOL_NEI; ++j) {
        int p = ib[j];
        m = fmaxf(m, feat[((size_t)b * Vin + p) * C + c]);
    }
    out[((size_t)b * Vout + v) * outStride + c] = m;
}

// ---------------------------------------------------------------------------
// Global max over vertices, then the tiny FC head.
// ---------------------------------------------------------------------------
__global__ void gmax_kernel(const float* __restrict__ f, int V, int C,
                            float* __restrict__ g)
{
    int c = blockIdx.x * blockDim.x + threadIdx.x;
    int b = blockIdx.y;
    if (c >= C) return;
    float m = NEG_BIG;
    for (int v = 0; v < V; ++v) m = fmaxf(m, f[((size_t)b * V + v) * C + c]);
    g[(size_t)b * C + c] = m;
}

__global__ void fc1_kernel(const float* __restrict__ g,
                           const float* __restrict__ w,
                           const float* __restrict__ bias,
                           const float* __restrict__ bng,
                           const float* __restrict__ bnb,
                           const float* __restrict__ bnm,
                           const float* __restrict__ bnv,
                           float* __restrict__ h)
{
    int c = blockIdx.x * blockDim.x + threadIdx.x;
    int b = blockIdx.y;
    if (c >= 256) return;
    float acc = bias[c];
    const float* gb = g + (size_t)b * 1024;
    for (int i = 0; i < 1024; ++i) acc = fmaf(gb[i], w[(size_t)i * 256 + c], acc);
    acc = (acc - bnm[c]) / sqrtf(bnv[c] + 1e-5f) * bng[c] + bnb[c];
    h[(size_t)b * 256 + c] = fmaxf(acc, 0.0f);
}

__global__ void fc2_kernel(const float* __restrict__ h,
                           const float* __restrict__ w,
                           const float* __restrict__ bias,
                           float* __restrict__ out)
{
    int c = blockIdx.x * blockDim.x + threadIdx.x;
    int b = blockIdx.y;
    if (c >= 40) return;
    float acc = bias[c];
    const float* hb = h + (size_t)b * 256;
    for (int i = 0; i < 256; ++i) acc = fmaf(hb[i], w[(size_t)i * 40 + c], acc);
    out[(size_t)b * 40 + c] = acc;
}

// ---------------------------------------------------------------------------
// Host orchestration
// ---------------------------------------------------------------------------
extern "C" void kernel_launch(void* const* d_in, const int* in_sizes, int n_in,
                              void* d_out, int out_size, void* d_ws, size_t ws_size,
                              hipStream_t stream)
{
    (void)in_sizes; (void)n_in; (void)out_size; (void)ws_size;
    const int B = 4;
    const float* verts = (const float*)d_in[0];
    const float* surf_dirs = (const float*)d_in[1];
    const float *W[9], *Bi[9], *D[9];
    for (int i = 0; i < 9; ++i) {
        W[i]  = (const float*)d_in[2 + 3 * i + 0];
        Bi[i] = (const float*)d_in[2 + 3 * i + 1];
        D[i]  = (const float*)d_in[2 + 3 * i + 2];
    }
    const float* fc1_w = (const float*)d_in[29];
    const float* fc1_b = (const float*)d_in[30];
    const float* bn_g  = (const float*)d_in[31];
    const float* bn_b  = (const float*)d_in[32];
    const float* bn_m  = (const float*)d_in[33];
    const float* bn_v  = (const float*)d_in[34];
    const float* fc2_w = (const float*)d_in[35];
    const float* fc2_b = (const float*)d_in[36];

    char* p = (char*)d_ws;
    auto alloc = [&](size_t bytes) -> void* {
        void* r = (void*)p;
        p += (bytes + 255) & ~(size_t)255;
        return r;
    };
    float* sd   = (float*)alloc((size_t)3 * 4096 * sizeof(float));
    int*   idxb = (int*)  alloc((size_t)B * 4096 * NEI * sizeof(int));
    float* L0   = (float*)alloc((size_t)B * 4096 * 64  * sizeof(float));
    float* L1   = (float*)alloc((size_t)B * 2048 * 128 * sizeof(float));
    float* L2   = (float*)alloc((size_t)B * 1024 * 192 * sizeof(float));
    float* L3   = (float*)alloc((size_t)B * 512  * 256 * sizeof(float));
    float* L4   = (float*)alloc((size_t)B * 256  * 288 * sizeof(float));
    float* FM11 = (float*)alloc((size_t)B * 256  * 1024 * sizeof(float));
    float* fout = (float*)alloc((size_t)B * 256  * 5120 * sizeof(float)); // max GEMM out
    float* g    = (float*)alloc((size_t)B * 1024 * sizeof(float));
    float* h    = (float*)alloc((size_t)B * 256  * sizeof(float));

    auto qball = [&](float radius, int Kq, int npts, int nq) {
        dim3 grid((nq + 127) / 128, B);
        query_ball_kernel<<<grid, 128, 0, stream>>>(verts, npts, nq,
                                                    radius * radius, Kq, idxb);
    };
    auto conv = [&](int Vl, const float* feat, int featStride, int cin,
                    const float* w, const float* bias, const float* dirs,
                    float* outB, int outStride, int outOff, int relu, int o) {
        const int N = 5 * o;
        const int strips = N / 64;               // full 16x64 strips
        const int tailTiles = (N - strips * 64) / 16;
        const int rowsTiles = (B * Vl) / 16;
        if (strips > 0) {
            dim3 grid((strips + 3) / 4, rowsTiles);
            wmma_gemm_bias_wide<<<grid, 128, 0, stream>>>(
                feat, featStride, w, N, bias, fout, cin, strips);
        }
        if (tailTiles > 0) {
            dim3 grid((tailTiles + 3) / 4, rowsTiles);
            wmma_gemm_bias_tile<<<grid, 128, 0, stream>>>(
                feat, featStride, w, N, bias, fout, cin, strips * 64, tailTiles);
        }
        const int Md = 4 * o;
        norm_dirs_kernel<<<(Md + 127) / 128, 128, 0, stream>>>(dirs, sd, Md);
        const int bs = (o > 32) ? 256 : 32;
        conv_agg_kernel<true><<<dim3(Vl, B), bs, 0, stream>>>(
            idxb, verts, sd, fout, outB, Vl, o, outStride, outOff, relu);
    };
    auto pool = [&](float radius, int Vin, int Cin, const float* inB,
                    float* outB, int outStride) {
        int Vout = Vin / 2;
        qball(radius, POOL_NEI, Vin, Vout);
        pool_max_kernel<<<dim3((Cin + 127) / 128, Vout, B), 128, 0, stream>>>(
            inB, Cin, Vin, idxb, outB, outStride, Vout);
    };

    // ---- Level 0 (V=4096) ----
    qball(0.2f, NEI, 4096, 4096);
    norm_dirs_kernel<<<1, 128, 0, stream>>>(surf_dirs, sd, 128);
    conv_agg_kernel<false><<<dim3(4096, B), 32, 0, stream>>>(
        idxb, verts, sd, nullptr, L0, 4096, 32, 64, 0, 1);        // fm0
    conv(4096, L0, 64, 32, W[0], Bi[0], D[0], L0, 64, 32, 1, 32); // fm1
    pool(0.2f, 4096, 64, L0, L1, 128);

    // ---- Level 1 (V=2048) ----
    qball(0.4f, NEI, 2048, 2048);
    conv(2048, L1, 128, 64, W[1], Bi[1], D[1], L1, 128, 64, 1, 32); // fm3
    conv(2048, L1, 128, 96, W[2], Bi[2], D[2], L1, 128, 96, 1, 32); // fm4
    pool(0.4f, 2048, 128, L1, L2, 192);

    // ---- Level 2 (V=1024) ----
    qball(0.6f, NEI, 1024, 1024);
    conv(1024, L2, 192, 128, W[3], Bi[3], D[3], L2, 192, 128, 1, 32); // fm6
    conv(1024, L2, 192, 160, W[4], Bi[4], D[4], L2, 192, 160, 1, 32); // fm7
    pool(0.6f, 1024, 192, L2, L3, 256);

    // ---- Level 3 (V=512) ----
    qball(0.8f, NEI, 512, 512);
    conv(512, L3, 256, 192, W[5], Bi[5], D[5], L3, 256, 192, 1, 32); // fm8
    conv(512, L3, 256, 224, W[6], Bi[6], D[6], L3, 256, 224, 1, 32); // fm9
    pool(0.8f, 512, 256, L3, L4, 288);

    // ---- Level 4 (V=256) ----
    qball(1.0f, NEI, 256, 256);
    conv(256, L4, 288, 256, W[7], Bi[7], D[7], L4, 288, 256, 1, 32);   // fm10
    conv(256, L4, 288, 288, W[8], Bi[8], D[8], FM11, 1024, 0, 0, 1024); // fm11

    // ---- Head ----
    gmax_kernel<<<dim3(8, B), 128, 0, stream>>>(FM11, 256, 1024, g);
    fc1_kernel<<<dim3(2, B), 128, 0, stream>>>(g, fc1_w, fc1_b, bn_g, bn_b,
                                               bn_m, bn_v, h);
    fc2_kernel<<<dim3(1, B), 64, 0, stream>>>(h, fc2_w, fc2_b, (float*)d_out);
}